// SubsampleGroup_60610578481795
// MI455X (gfx1250) — compile-verified
//
#include <hip/hip_runtime.h>

#define BQ   8
#define NQ   16384
#define CQ   128
#define MQ   256      // NUM_GROUPS
#define NSQ  32       // GROUP_SIZE
#define R2Q  0.01f    // RADIUS^2

// output tuple flat offsets
#define OFF_GP  0                            // grouped_p  B*3*M*NS = 196608
#define OFF_CP  (BQ*3*MQ*NSQ)                // center_p   B*M*3    = 6144
#define OFF_FJ  (OFF_CP + BQ*MQ*3)           // fj         B*C*M*NS = 8388608
#define OFF_CX  (OFF_FJ + BQ*CQ*MQ*NSQ)      // center_x   B*C*M    = 262144

typedef __attribute__((ext_vector_type(2))) float v2f;
typedef __attribute__((ext_vector_type(8))) float v8f;

// ---------------------------------------------------------------------------
// 1) Furthest point sampling: one block per batch. Point coords and dist[]
//    live entirely in VGPRs (16 elems/thread x 1024 threads); per-iteration
//    argmax = 5-step wave32 shuffle reduce + one LDS hop over 32 waves +
//    second shuffle reduce (2 barriers/iter). Tie-breaking matches
//    jnp.argmax (smallest index on equal max).
// ---------------------------------------------------------------------------
__global__ __launch_bounds__(1024)
void fps_kernel(const float* __restrict__ p, int* __restrict__ idx,
                float* __restrict__ center_p)
{
    const int b    = blockIdx.x;
    const int tid  = threadIdx.x;
    const int lane = tid & 31;
    const int wave = tid >> 5;
    const float* pb = p + (size_t)b * NQ * 3;

    __shared__ float s_val[32];
    __shared__ int   s_arg[32];
    __shared__ float s_cur[3];

    float px[16], py[16], pz[16], dist[16];
#pragma unroll
    for (int k = 0; k < 16; ++k) {
        const int n = tid + k * 1024;
        px[k] = pb[n * 3 + 0];
        py[k] = pb[n * 3 + 1];
        pz[k] = pb[n * 3 + 2];
        dist[k] = 1e10f;
    }

    if (tid == 0) {
        idx[b * MQ] = 0;
        const float cx = pb[0], cy = pb[1], cz = pb[2];
        s_cur[0] = cx; s_cur[1] = cy; s_cur[2] = cz;
        float* cp = center_p + (size_t)b * MQ * 3;
        cp[0] = cx; cp[1] = cy; cp[2] = cz;
    }
    __syncthreads();

    for (int m = 1; m < MQ; ++m) {
        const float cx = s_cur[0], cy = s_cur[1], cz = s_cur[2];
        float lmax = -1.0f; int larg = 0;
#pragma unroll
        for (int k = 0; k < 16; ++k) {
            const float dx = px[k] - cx;
            const float dy = py[k] - cy;
            const float dz = pz[k] - cz;
            const float d  = dx * dx + dy * dy + dz * dz;
            const float nd = fminf(dist[k], d);
            dist[k] = nd;
            if (nd > lmax) { lmax = nd; larg = tid + k * 1024; }  // strict > keeps first
        }
        // wave32 argmax reduce
#pragma unroll
        for (int off = 16; off > 0; off >>= 1) {
            const float ov = __shfl_xor(lmax, off);
            const int   oa = __shfl_xor(larg, off);
            if (ov > lmax || (ov == lmax && oa < larg)) { lmax = ov; larg = oa; }
        }
        if (lane == 0) { s_val[wave] = lmax; s_arg[wave] = larg; }
        __syncthreads();
        if (wave == 0) {
            float v = s_val[lane]; int a = s_arg[lane];
#pragma unroll
            for (int off = 16; off > 0; off >>= 1) {
                const float ov = __shfl_xor(v, off);
                const int   oa = __shfl_xor(a, off);
                if (ov > v || (ov == v && oa < a)) { v = ov; a = oa; }
            }
            if (lane == 0) {
                idx[b * MQ + m] = a;
                const float nx = pb[a * 3 + 0];
                const float ny = pb[a * 3 + 1];
                const float nz = pb[a * 3 + 2];
                s_cur[0] = nx; s_cur[1] = ny; s_cur[2] = nz;
                float* cp = center_p + ((size_t)b * MQ + m) * 3;
                cp[0] = nx; cp[1] = ny; cp[2] = nz;
            }
        }
        __syncthreads();
    }
}

// ---------------------------------------------------------------------------
// 2) Ball query via V_WMMA_F32_16X16X4_F32:
//    d2[m][n] = |c_m|^2 - 2 c_m.s_n + |s_n|^2 in one WMMA per 16x16 tile.
//    A row m = (-2cx,-2cy,-2cz, 1): lanes 0-15 hold K=0,1; lanes 16-31 K=2,3.
//    B col n = (sx, sy, sz, |s|^2) with the matching K split.
//    C accumulator preloaded with |c_m|^2 (VGPR r -> row r + 8*(lane/16)).
//    In-radius mask: 8 full-EXEC ballots (VGPR r ballot: bits[15:0] = row r,
//    bits[31:16] = row r+8). Ballot results are wave-uniform SGPRs; each
//    center lane picks its row mask via a branch-free cndmask select chain
//    (no LDS, no barriers in the hot loop), then bit-scans it in index order
//    (ctz), matching the CUDA ball_query "first nsample in index order, pad
//    with first hit" rule.
// ---------------------------------------------------------------------------
__global__ __launch_bounds__(32)
void ballq_kernel(const float* __restrict__ p, const float* __restrict__ center_p,
                  int* __restrict__ nidx)
{
    const int blk  = blockIdx.x;          // B * (MQ/16) = 128
    const int b    = blk >> 4;
    const int m0   = (blk & 15) * 16;
    const int lane = threadIdx.x;
    const int col  = lane & 15;
    const int half = lane >> 4;
    const int r8   = lane & 7;
    const float* pb = p + (size_t)b * NQ * 3;

    __shared__ float s_c[16][3];
    __shared__ float s_csq[16];

    if (lane < 16) {
        const float* cp = center_p + ((size_t)b * MQ + m0 + lane) * 3;
        const float cx = cp[0], cy = cp[1], cz = cp[2];
        s_c[lane][0] = cx; s_c[lane][1] = cy; s_c[lane][2] = cz;
        s_csq[lane]  = cx * cx + cy * cy + cz * cz;
    }
    __syncthreads();

    v2f a;
    if (half == 0) { a.x = -2.0f * s_c[col][0]; a.y = -2.0f * s_c[col][1]; }
    else           { a.x = -2.0f * s_c[col][2]; a.y = 1.0f; }

    v8f cacc;
#pragma unroll
    for (int r = 0; r < 8; ++r) cacc[r] = s_csq[r + 8 * half];

    int cnt = 0, first = -1;
    int* const row_out = nidx + ((b * MQ + m0 + col) * NSQ);

    for (int n0 = 0; n0 < NQ; n0 += 16) {
        // speculative stream prefetch (global_prefetch_b8); OOB is dropped
        __builtin_prefetch(pb + (size_t)(n0 + 128) * 3, 0, 0);

        const int j = n0 + col;
        const float sx = pb[j * 3 + 0], sy = pb[j * 3 + 1], sz = pb[j * 3 + 2];
        v2f bb;
        if (half == 0) { bb.x = sx; bb.y = sy; }
        else           { bb.x = sz; bb.y = sx * sx + sy * sy + sz * sz; }

        // 8 args: (neg_a, A, neg_b, B, c_mod, C, reuse_a, reuse_b)
        const v8f d = __builtin_amdgcn_wmma_f32_16x16x4_f32(
            false, a, false, bb, (short)0, cacc, false, false);

        // full-EXEC ballots: wave-uniform masks, one per accumulator VGPR;
        // bits[15:0] = row r (lanes 0-15), bits[31:16] = row r+8 (lanes 16-31)
        unsigned bal[8];
#pragma unroll
        for (int r = 0; r < 8; ++r) bal[r] = (unsigned)__ballot(d[r] < R2Q);

        // branch-free per-lane row-mask select (SGPR -> VGPR cndmask chain)
        unsigned m32 = bal[0];
#pragma unroll
        for (int r = 1; r < 8; ++r) m32 = (r8 == r) ? bal[r] : m32;
        unsigned mask = (lane < 8) ? (m32 & 0xFFFFu) : (m32 >> 16);

        if (lane < 16 && cnt < NSQ) {
            while (mask && cnt < NSQ) {
                const int jj = __builtin_ctz(mask);
                mask &= mask - 1;
                const int ng = n0 + jj;
                row_out[cnt] = ng;
                if (first < 0) first = ng;
                ++cnt;
            }
        }

        const int done = (lane < 16) ? (cnt >= NSQ) : 1;
        if (__all(done)) break;   // uniform exit keeps EXEC all-ones at WMMA
    }

    if (lane < 16) {
        if (first < 0) first = 0;
        for (int s = cnt; s < NSQ; ++s) row_out[s] = first;
    }
}

// ---------------------------------------------------------------------------
// 3) grouped_p[b][d][m][s] = p[b][nidx[b][m][s]][d] - center_p[b][m][d]
// ---------------------------------------------------------------------------
__global__ __launch_bounds__(256)
void group_p_kernel(const float* __restrict__ p, const int* __restrict__ nidx,
                    const float* __restrict__ center_p, float* __restrict__ out)
{
    const int g = blockIdx.x * blockDim.x + threadIdx.x;
    if (g >= BQ * 3 * MQ * NSQ) return;
    const int s = g & (NSQ - 1);
    const int m = (g / NSQ) & (MQ - 1);
    const int d = (g / (NSQ * MQ)) % 3;
    const int b = g / (NSQ * MQ * 3);
    const int n = nidx[(b * MQ + m) * NSQ + s];
    const float pv = p[((size_t)b * NQ + n) * 3 + d];
    const float cv = center_p[((size_t)b * MQ + m) * 3 + d];
    out[g] = pv - cv;
}

// ---------------------------------------------------------------------------
// 4) fj[b][c][m][s] = x[b][c][nidx[b][m][s]]  (coalesced stores over s)
// ---------------------------------------------------------------------------
__global__ __launch_bounds__(256)
void gather_fj_kernel(const float* __restrict__ x, const int* __restrict__ nidx,
                      float* __restrict__ fj)
{
    const int b = blockIdx.x / MQ;
    const int m = blockIdx.x % MQ;
    __shared__ int s_n[NSQ];
    if (threadIdx.x < NSQ)
        s_n[threadIdx.x] = nidx[(b * MQ + m) * NSQ + threadIdx.x];
    __syncthreads();
    const int s  = threadIdx.x & (NSQ - 1);
    const int c0 = threadIdx.x / NSQ;     // 0..7
    const int n  = s_n[s];
    for (int c = c0; c < CQ; c += 8) {
        const float v = x[((size_t)(b * CQ + c)) * NQ + n];
        fj[(((size_t)(b * CQ + c)) * MQ + m) * NSQ + s] = v;
    }
}

// ---------------------------------------------------------------------------
// 5) center_x[b][c][m] = x[b][c][idx[b][m]]
// ---------------------------------------------------------------------------
__global__ __launch_bounds__(256)
void center_x_kernel(const float* __restrict__ x, const int* __restrict__ idx,
                     float* __restrict__ cx)
{
    const int g = blockIdx.x * blockDim.x + threadIdx.x;
    if (g >= BQ * CQ * MQ) return;
    const int m = g & (MQ - 1);
    const int c = (g / MQ) & (CQ - 1);
    const int b = g / (MQ * CQ);
    const int n = idx[b * MQ + m];
    cx[g] = x[((size_t)(b * CQ + c)) * NQ + n];
}

extern "C" void kernel_launch(void* const* d_in, const int* in_sizes, int n_in,
                              void* d_out, int out_size, void* d_ws, size_t ws_size,
                              hipStream_t stream)
{
    const float* p = (const float*)d_in[0];   // [B,N,3]
    const float* x = (const float*)d_in[1];   // [B,C,N]
    float* out       = (float*)d_out;
    float* grouped_p = out + OFF_GP;
    float* center_p  = out + OFF_CP;
    float* fj        = out + OFF_FJ;
    float* center_x  = out + OFF_CX;

    int* idx  = (int*)d_ws;                   // [B,M]
    int* nidx = idx + BQ * MQ;                // [B,M,NS]

    fps_kernel<<<BQ, 1024, 0, stream>>>(p, idx, center_p);
    ballq_kernel<<<BQ * (MQ / 16), 32, 0, stream>>>(p, center_p, nidx);
    group_p_kernel<<<(BQ * 3 * MQ * NSQ + 255) / 256, 256, 0, stream>>>(
        p, nidx, center_p, grouped_p);
    gather_fj_kernel<<<BQ * MQ, 256, 0, stream>>>(x, nidx, fj);
    center_x_kernel<<<(BQ * CQ * MQ + 255) / 256, 256, 0, stream>>>(x, idx, center_x);
}